// CGNN_16827681865778
// MI455X (gfx1250) — compile-verified
//
#include <hip/hip_runtime.h>

typedef __attribute__((ext_vector_type(2))) float v2f;
typedef __attribute__((ext_vector_type(4))) float v4f;
typedef __attribute__((ext_vector_type(8))) float v8f;

#define BSZ    16384
#define DU     20
#define NROWS  (BSZ * DU)       // 327680
#define NTILES (NROWS / 16)     // 20480
#define WAVES  4
#define THREADS (WAVES * 32)
#define NBLOCKS (NTILES / WAVES) // 5120

// output segment offsets (floats)
#define F1_OFF 0
#define G1_OFF 327680
#define F2_OFF 33095680
#define G2_OFF 34734080

// LDS strides (all even so v2f/b64 loads are 8B aligned)
#define S1  4    // K-stride for W1 (K padded 3->4)
#define S16 18   // K-stride for K=16 weights
#define S32 34   // K-stride for K=32 weights
#define HST 34   // hbuf row stride (16 rows x up to 32 cols)
#define CST 132  // cbuf row stride (16 rows x up to 130 cols)

// LDS layout (floats)
#define OFF_WT1A 0
#define OFF_WT2A (OFF_WT1A + 16 * S1)     // 64
#define OFF_WT3A (OFF_WT2A + 32 * S16)    // 640
#define OFF_WT4A (OFF_WT3A + 16 * S32)    // 1184
#define OFF_B1A  (OFF_WT4A + 16 * S16)    // 1472
#define OFF_B2A  (OFF_B1A + 16)
#define OFF_B3A  (OFF_B2A + 32)
#define OFF_B4A  (OFF_B3A + 16)
#define OFF_WT1B (OFF_B4A + 16)           // 1552
#define OFF_WT2B (OFF_WT1B + 16 * S1)
#define OFF_WT3B (OFF_WT2B + 32 * S16)
#define OFF_WT4B (OFF_WT3B + 16 * S32)
#define OFF_B1B  (OFF_WT4B + 130 * S16)   // 5076
#define OFF_B2B  (OFF_B1B + 16)
#define OFF_B3B  (OFF_B2B + 32)
#define OFF_B4B  (OFF_B3B + 16)
#define OFF_WAVE (OFF_B4B + 130)          // 5270
#define PER_WAVE (16 * HST + 16 * CST)    // 2656
#define LDS_TOTAL (OFF_WAVE + WAVES * PER_WAVE)  // 15894 floats = 63576 B

__device__ __forceinline__ void lds_fence() {
#if __has_builtin(__builtin_amdgcn_s_wait_dscnt)
  __builtin_amdgcn_s_wait_dscnt(0);
  asm volatile("" ::: "memory");
#else
  asm volatile("s_wait_dscnt 0" ::: "memory");
#endif
}

__device__ __forceinline__ v8f wmma4(v2f a, v2f b, v8f c) {
  // D = A(16x4 f32) * B(4x16 f32) + C(16x16 f32)
  return __builtin_amdgcn_wmma_f32_16x16x4_f32(
      /*neg_a=*/false, a, /*neg_b=*/false, b,
      /*c_mod=*/(short)0, c, /*reuse_a=*/false, /*reuse_b=*/false);
}

// Load weight matrix (K,N) row-major from global, store transposed [n*S + k] in LDS.
__device__ __forceinline__ void load_wt(float* dst, const float* __restrict__ w,
                                        int K, int N, int S, int tid) {
  for (int i = tid; i < K * N; i += THREADS) {
    int k = i / N, n = i - k * N;
    dst[n * S + k] = w[i];
  }
}

// Run one 4-layer MLP on a 16-row tile. Input A-fragment ax (from x_local).
// Final layer output staged compactly into cbuf (16 x nt4*16 cols, clipped to CST).
__device__ __forceinline__ void run_net(float* lds, float* hbuf, float* cbuf,
                                        v2f ax, int mq, int hq,
                                        int OW1, int OW2, int OW3, int OW4,
                                        int OB1, int OB2, int OB3, int OB4,
                                        int nt4) {
  // ---- layer 1: (16x4) @ (4x16) -> 16x16, relu ----
  {
    float bv = lds[OB1 + mq];
    v8f acc;
#pragma unroll
    for (int i = 0; i < 8; i++) acc[i] = bv;
    v2f bf = *(const v2f*)(lds + OW1 + mq * S1 + 2 * hq);
    acc = wmma4(ax, bf, acc);
#pragma unroll
    for (int i = 0; i < 8; i++) {
      float v = acc[i];
      hbuf[(i + 8 * hq) * HST + mq] = (v > 0.f) ? v : 0.f;
    }
  }
  lds_fence();
  // ---- layer 2: (16x16) @ (16x32) -> 16x32, relu ----
  {
    v2f afr[4];
#pragma unroll
    for (int k = 0; k < 4; k++)
      afr[k] = *(const v2f*)(hbuf + mq * HST + 4 * k + 2 * hq);
#pragma unroll
    for (int t = 0; t < 2; t++) {
      int n = 16 * t + mq;
      float bv = lds[OB2 + n];
      v8f acc;
#pragma unroll
      for (int i = 0; i < 8; i++) acc[i] = bv;
#pragma unroll
      for (int k = 0; k < 4; k++) {
        v2f bf = *(const v2f*)(lds + OW2 + n * S16 + 4 * k + 2 * hq);
        acc = wmma4(afr[k], bf, acc);
      }
#pragma unroll
      for (int i = 0; i < 8; i++) {
        float v = acc[i];
        hbuf[(i + 8 * hq) * HST + n] = (v > 0.f) ? v : 0.f;
      }
    }
  }
  lds_fence();
  // ---- layer 3: (16x32) @ (32x16) -> 16x16, relu ----
  {
    v2f afr[8];
#pragma unroll
    for (int k = 0; k < 8; k++)
      afr[k] = *(const v2f*)(hbuf + mq * HST + 4 * k + 2 * hq);
    float bv = lds[OB3 + mq];
    v8f acc;
#pragma unroll
    for (int i = 0; i < 8; i++) acc[i] = bv;
#pragma unroll
    for (int k = 0; k < 8; k++) {
      v2f bf = *(const v2f*)(lds + OW3 + mq * S32 + 4 * k + 2 * hq);
      acc = wmma4(afr[k], bf, acc);
    }
    lds_fence();  // ensure layer-3 input reads done before overwrite below
#pragma unroll
    for (int i = 0; i < 8; i++) {
      float v = acc[i];
      hbuf[(i + 8 * hq) * HST + mq] = (v > 0.f) ? v : 0.f;
    }
  }
  lds_fence();
  // ---- layer 4: (16x16) @ (16 x nt4*16) -> staged to cbuf, no relu ----
  {
    v2f afr[4];
#pragma unroll
    for (int k = 0; k < 4; k++)
      afr[k] = *(const v2f*)(hbuf + mq * HST + 4 * k + 2 * hq);
    for (int t = 0; t < nt4; t++) {
      int n = 16 * t + mq;
      float bv = lds[OB4 + n];  // n may exceed valid range for t=8 (garbage cols, never read)
      v8f acc;
#pragma unroll
      for (int i = 0; i < 8; i++) acc[i] = bv;
#pragma unroll
      for (int k = 0; k < 4; k++) {
        v2f bf = *(const v2f*)(lds + OW4 + n * S16 + 4 * k + 2 * hq);
        acc = wmma4(afr[k], bf, acc);
      }
      if (n < CST) {
#pragma unroll
        for (int i = 0; i < 8; i++) cbuf[(i + 8 * hq) * CST + n] = acc[i];
      }
    }
  }
  lds_fence();
}

__global__ __launch_bounds__(THREADS)
void cgnn_mlp_scatter_kernel(
    const float* __restrict__ x,
    const float* __restrict__ w1a, const float* __restrict__ b1a,
    const float* __restrict__ w2a, const float* __restrict__ b2a,
    const float* __restrict__ w3a, const float* __restrict__ b3a,
    const float* __restrict__ w4a, const float* __restrict__ b4a,
    const float* __restrict__ w1b, const float* __restrict__ b1b,
    const float* __restrict__ w2b, const float* __restrict__ b2b,
    const float* __restrict__ w3b, const float* __restrict__ b3b,
    const float* __restrict__ w4b, const float* __restrict__ b4b,
    float* __restrict__ out) {
  __shared__ float lds[LDS_TOTAL];
  const int tid = threadIdx.x;

  // ---- stage all weights transposed (+biases) into LDS ----
  load_wt(lds + OFF_WT1A, w1a, 3, 16, S1, tid);
  load_wt(lds + OFF_WT2A, w2a, 16, 32, S16, tid);
  load_wt(lds + OFF_WT3A, w3a, 32, 16, S32, tid);
  load_wt(lds + OFF_WT4A, w4a, 16, 16, S16, tid);
  load_wt(lds + OFF_WT1B, w1b, 3, 16, S1, tid);
  load_wt(lds + OFF_WT2B, w2b, 16, 32, S16, tid);
  load_wt(lds + OFF_WT3B, w3b, 32, 16, S32, tid);
  load_wt(lds + OFF_WT4B, w4b, 16, 130, S16, tid);
  for (int i = tid; i < 16; i += THREADS) {
    lds[OFF_WT1A + i * S1 + 3] = 0.f;  // pad K=3 row of layer-1 weights
    lds[OFF_WT1B + i * S1 + 3] = 0.f;
    lds[OFF_B1A + i] = b1a[i];
    lds[OFF_B3A + i] = b3a[i];
    lds[OFF_B4A + i] = b4a[i];
    lds[OFF_B1B + i] = b1b[i];
    lds[OFF_B3B + i] = b3b[i];
  }
  for (int i = tid; i < 32; i += THREADS) {
    lds[OFF_B2A + i] = b2a[i];
    lds[OFF_B2B + i] = b2b[i];
  }
  for (int i = tid; i < 130; i += THREADS) lds[OFF_B4B + i] = b4b[i];
  __syncthreads();

  const int lane = tid & 31;
  const int wv = tid >> 5;
  const int mq = lane & 15;  // row (for A) / col (for B,C,D) index
  const int hq = lane >> 4;  // half selector
  float* hbuf = lds + OFF_WAVE + wv * PER_WAVE;
  float* cbuf = hbuf + 16 * HST;

  const int tile = blockIdx.x * WAVES + wv;
  if (tile >= NTILES) return;

  float* f1 = out + F1_OFF;
  float* g1 = out + G1_OFF;
  float* f2 = out + F2_OFF;
  float* g2 = out + G2_OFF;

  // ---- A-fragment for layer 1: x_local rows (neighbors u-1, u, u+1, pad) ----
  const int r = tile * 16 + mq;
  const int bb = r / 20;
  const int uu = r - bb * 20;
  v2f ax;
  if (hq == 0) {
    int um = (uu == 0) ? 19 : uu - 1;
    ax.x = x[bb * 20 + um];   // K=0
    ax.y = x[bb * 20 + uu];   // K=1
  } else {
    int up = (uu == 19) ? 0 : uu + 1;
    ax.x = x[bb * 20 + up];   // K=2
    ax.y = 0.f;               // K=3 (pad)
  }

  // ==================== net a ====================
  run_net(lds, hbuf, cbuf, ax, mq, hq,
          OFF_WT1A, OFF_WT2A, OFF_WT3A, OFF_WT4A,
          OFF_B1A, OFF_B2A, OFF_B3A, OFF_B4A, 1);

  // f1: out1[:,0]
  if (lane < 16) f1[tile * 16 + lane] = cbuf[lane * CST];

  // g1: build full 100-col rows (rotation + zero fill) and stream out
  for (int idx = lane; idx < 16 * 25; idx += 32) {
    int mm = idx / 25;
    int q = idx - mm * 25;
    int rr = tile * 16 + mm;
    int u2 = rr % 20;
    int s = ((u2 + 19) % 20) * 5;  // span start = 5*(u-1) mod 100
    const float* src = cbuf + mm * CST + 1;
    v4f v4;
#pragma unroll
    for (int e = 0; e < 4; e++) {
      int c = q * 4 + e;
      int rel = c - s;
      rel += (rel >> 31) & 100;
      float vv = 0.f;
      if (rel < 15) vv = src[rel];
      v4[e] = vv;
    }
    __builtin_nontemporal_store(
        v4, reinterpret_cast<v4f*>(g1 + (size_t)rr * 100 + q * 4));
  }

  // ==================== net b ====================
  run_net(lds, hbuf, cbuf, ax, mq, hq,
          OFF_WT1B, OFF_WT2B, OFF_WT3B, OFF_WT4B,
          OFF_B1B, OFF_B2B, OFF_B3B, OFF_B4B, 9);

  // f2: out2[:, :, 0:5] -> contiguous 80 values per tile
  for (int idx = lane; idx < 80; idx += 32) {
    int mm = idx / 5;
    int z = idx - mm * 5;
    f2[tile * 80 + idx] = cbuf[mm * CST + z];
  }

  // g2: 80 rows x 100 cols per tile, span of 25 starting at 5*(u-2) mod 100
  for (int idx = lane; idx < 80 * 25; idx += 32) {
    int row = idx / 25;        // 0..79 = mm*5 + z
    int q = idx - row * 25;
    int mm = row / 5;
    int z = row - mm * 5;
    int rr = tile * 16 + mm;
    int u2 = rr % 20;
    int s = ((u2 + 18) % 20) * 5;  // span start = 5*(u-2) mod 100
    const float* src = cbuf + mm * CST + 5 + z * 25;
    v4f v4;
#pragma unroll
    for (int e = 0; e < 4; e++) {
      int c = q * 4 + e;
      int rel = c - s;
      rel += (rel >> 31) & 100;
      float vv = 0.f;
      if (rel < 25) vv = src[rel];
      v4[e] = vv;
    }
    __builtin_nontemporal_store(
        v4,
        reinterpret_cast<v4f*>(g2 + (size_t)(tile * 80 + row) * 100 + q * 4));
  }
}

extern "C" void kernel_launch(void* const* d_in, const int* in_sizes, int n_in,
                              void* d_out, int out_size, void* d_ws,
                              size_t ws_size, hipStream_t stream) {
  const float* x   = (const float*)d_in[0];
  const float* w1a = (const float*)d_in[1];
  const float* b1a = (const float*)d_in[2];
  const float* w2a = (const float*)d_in[3];
  const float* b2a = (const float*)d_in[4];
  const float* w3a = (const float*)d_in[5];
  const float* b3a = (const float*)d_in[6];
  const float* w4a = (const float*)d_in[7];
  const float* b4a = (const float*)d_in[8];
  const float* w1b = (const float*)d_in[9];
  const float* b1b = (const float*)d_in[10];
  const float* w2b = (const float*)d_in[11];
  const float* b2b = (const float*)d_in[12];
  const float* w3b = (const float*)d_in[13];
  const float* b3b = (const float*)d_in[14];
  const float* w4b = (const float*)d_in[15];
  const float* b4b = (const float*)d_in[16];

  cgnn_mlp_scatter_kernel<<<NBLOCKS, THREADS, 0, stream>>>(
      x, w1a, b1a, w2a, b2a, w3a, b3a, w4a, b4a,
      w1b, b1b, w2b, b2b, w3b, b3b, w4b, b4b, (float*)d_out);
}